// AlphaNet_25254407701112
// MI455X (gfx1250) — compile-verified
//
#include <hip/hip_runtime.h>

// CDNA5 (gfx1250) periodic radius-graph (OCP/AlphaNet-style neighbor list).
//
// Roofline: outputs = 453 MB of f32 stores -> ~19.5 us at 23.3 TB/s HBM.
// Compute (28.3M squared distances) ~0.25 GFLOP -> negligible. The kernel is
// store-bound, so pass 3 uses 128-bit non-temporal stores (write-only stream
// larger than the 192 MB L2). Distances are evaluated on the matrix pipe with
// V_WMMA_F32_16X16X4_F32 via the Gram identity d^2 = |p|^2 - 2 p.q + |q|^2
// (A = [-2p, 1], B = [q, |q|^2], C preloaded with |p|^2); LDS handles the
// exact stable top-32 selection. 256 WGs x 1024 threads = 8192 wave32.

typedef __attribute__((ext_vector_type(2))) float v2f;
typedef __attribute__((ext_vector_type(4))) float v4f;
typedef __attribute__((ext_vector_type(8))) float v8f;

#define BATCH    16
#define NATOM    256
#define NCELL    27
#define MCAND    (NATOM * NCELL)   // 6912 candidates per receiver
#define MAXNBR   32
#define RAD2     25.0f
#define TILES    (MCAND / 16)      // 432 WMMA column tiles
#define NTHREADS 1024              // 32 wave32 per workgroup
#define NWAVES   (NTHREADS / 32)
#define KEEPW    (MCAND / 32)      // 216 words of keep-bitmask per receiver

__global__ void zero_nimg_kernel(int* __restrict__ nimg) {
    if (threadIdx.x < BATCH) nimg[threadIdx.x] = 0;
}

__global__ __launch_bounds__(NTHREADS) void radius_graph_kernel(
    const float* __restrict__ pos,    // [B,N,3]
    const float* __restrict__ cell,   // [B,3,3]
    float* __restrict__ dist_out,     // [B,N,M]
    float* __restrict__ dvec_out,     // [B,N,M,3]
    int*   __restrict__ nimg)         // [B]
{
    __shared__ float    s_pos[NATOM][3];     // all atom positions of this batch
    __shared__ float    s_off[NCELL][3];     // 27 periodic image offsets
    __shared__ float    s_psq[16];           // |p_i|^2 for the 16 receivers
    __shared__ int      s_cnt[16];           // per-receiver within-count
    __shared__ float    s_ld[16][256];       // per-receiver candidate d^2
    __shared__ int      s_lm[16][256];       // per-receiver candidate flat index m
    __shared__ unsigned s_keep[16][KEEPW];   // per-receiver keep bitmask over m

    const int tid = threadIdx.x;
    const int g   = blockIdx.x;        // 0..255
    const int b   = g >> 4;            // batch
    const int i0  = (g & 15) << 4;     // first of 16 receivers handled here

    // ---------------- setup: positions, image offsets, clears ----------------
    {
        if (tid < NATOM) {
            const float* pb = pos + (size_t)b * NATOM * 3;
            s_pos[tid][0] = pb[tid * 3 + 0];
            s_pos[tid][1] = pb[tid * 3 + 1];
            s_pos[tid][2] = pb[tid * 3 + 2];
        }
        if (tid < NCELL) {
            const int c = tid;
            const float n1 = (float)(c / 9) - 1.0f;
            const float n2 = (float)((c / 3) % 3) - 1.0f;
            const float n3 = (float)(c % 3) - 1.0f;
            const float* cb = cell + b * 9;
            for (int d = 0; d < 3; ++d)
                s_off[c][d] = n1 * cb[0 * 3 + d] + n2 * cb[1 * 3 + d] + n3 * cb[2 * 3 + d];
        }
        if (tid < 16) s_cnt[tid] = 0;
        for (int w = tid; w < 16 * KEEPW; w += NTHREADS)
            ((unsigned*)s_keep)[w] = 0u;
    }
    __syncthreads();
    if (tid < 16) {
        const float x = s_pos[i0 + tid][0];
        const float y = s_pos[i0 + tid][1];
        const float z = s_pos[i0 + tid][2];
        s_psq[tid] = x * x + y * y + z * z;
    }
    __syncthreads();

    // ---------------- pass 1: WMMA d^2 tiles + compact within-lists ----------
    const int wave = tid >> 5;
    const int lane = tid & 31;
    const int l15  = lane & 15;   // A-row index AND B/D-column index for this lane
    const int half = lane >> 4;

    // A (16x4 f32): row m = l15; lanes 0-15 hold K={0,1}, lanes 16-31 K={2,3}
    const float px = s_pos[i0 + l15][0];
    const float py = s_pos[i0 + l15][1];
    const float pz = s_pos[i0 + l15][2];
    v2f a;
    a.x = half ? (-2.0f * pz) : (-2.0f * px);   // branchless (cndmask)
    a.y = half ? 1.0f         : (-2.0f * py);

    for (int tile = wave; tile < TILES; tile += NWAVES) {
        const int m = tile * 16 + l15;     // candidate column for this lane
        const int j = m / NCELL;
        const int c = m - j * NCELL;
        const float qx = s_pos[j][0] + s_off[c][0];
        const float qy = s_pos[j][1] + s_off[c][1];
        const float qz = s_pos[j][2] + s_off[c][2];
        const float qsq = qx * qx + qy * qy + qz * qz;
        // B (4x16 f32): col n = l15; lanes 0-15 hold K={0,1}, lanes 16-31 K={2,3}
        v2f bm;
        bm.x = half ? qz  : qx;            // branchless: keep EXEC all-1s
        bm.y = half ? qsq : qy;

        // C preloaded with |p_row|^2: VGPR vr = row vr (lanes 0-15) / vr+8 (16-31)
        v8f acc;
#pragma unroll
        for (int vr = 0; vr < 8; ++vr) acc[vr] = s_psq[vr + 8 * half];

        acc = __builtin_amdgcn_wmma_f32_16x16x4_f32(
            /*neg_a=*/false, a, /*neg_b=*/false, bm,
            /*c_mod=*/(short)0, acc, /*reuse_a=*/false, /*reuse_b=*/false);

        // Each lane holds d^2 for 8 receiver rows at its own candidate m.
#pragma unroll
        for (int vr = 0; vr < 8; ++vr) {
            const int   rrow = vr + 8 * half;
            const float ds   = acc[vr];
            const bool  self = (j == (i0 + rrow)) && (c == 13); // exact self image
            if (ds <= RAD2 && ds > 1e-4f && !self) {
                const int slot = atomicAdd(&s_cnt[rrow], 1);
                if (slot < 256) { s_ld[rrow][slot] = ds; s_lm[rrow][slot] = m; }
            }
        }
    }
    __syncthreads();

    // ---------------- pass 2: exact (d^2, m)-lexicographic top-32 ------------
    for (int r = 0; r < 16; ++r) {
        int L = s_cnt[r]; if (L > 256) L = 256;
        if (tid < L) {
            const float d = s_ld[r][tid];
            const int   m = s_lm[r][tid];
            int rank = 0;
            for (int e = 0; e < L; ++e) {
                const float de = s_ld[r][e];
                const int   me = s_lm[r][e];
                rank += (de < d) || (de == d && me < m);  // stable-sort tie-break
            }
            if (rank < MAXNBR)
                atomicOr(&s_keep[r][m >> 5], 1u << (m & 31));
        }
    }
    __syncthreads();
    if (tid < 16) {
        int L = s_cnt[tid]; if (L > 256) L = 256;
        atomicAdd(&nimg[b], (L < MAXNBR) ? L : MAXNBR);
    }

    // ------- pass 3: stream dense outputs with 128-bit non-temporal stores ---
    for (int r = 0; r < 16; ++r) {
        const int   i   = i0 + r;
        const float pix = s_pos[i][0], piy = s_pos[i][1], piz = s_pos[i][2];
        const size_t base = ((size_t)b * NATOM + i) * MCAND;
        for (int m0 = tid * 4; m0 < MCAND; m0 += NTHREADS * 4) {
            v4f dstv;
            v4f dv0, dv1, dv2;
            float dvbuf[12];
#pragma unroll
            for (int k = 0; k < 4; ++k) {
                const int m = m0 + k;
                const unsigned bit = (s_keep[r][m >> 5] >> (m & 31)) & 1u;
                float dst = 0.0f, dx = 0.0f, dy = 0.0f, dz = 0.0f;
                if (bit) {
                    const int j = m / NCELL;
                    const int c = m - j * NCELL;
                    // Reference's exact subtraction form for kept edges.
                    dx = pix - (s_pos[j][0] + s_off[c][0]);
                    dy = piy - (s_pos[j][1] + s_off[c][1]);
                    dz = piz - (s_pos[j][2] + s_off[c][2]);
                    dst = sqrtf(dx * dx + dy * dy + dz * dz);
                }
                dstv[k] = dst;
                dvbuf[3 * k + 0] = dx;
                dvbuf[3 * k + 1] = dy;
                dvbuf[3 * k + 2] = dz;
            }
            dv0 = (v4f){dvbuf[0], dvbuf[1], dvbuf[2],  dvbuf[3]};
            dv1 = (v4f){dvbuf[4], dvbuf[5], dvbuf[6],  dvbuf[7]};
            dv2 = (v4f){dvbuf[8], dvbuf[9], dvbuf[10], dvbuf[11]};

            __builtin_nontemporal_store(dstv, (v4f*)(dist_out + base + m0));
            v4f* vp = (v4f*)(dvec_out + (base + (size_t)m0) * 3);  // 16B aligned
            __builtin_nontemporal_store(dv0, vp + 0);
            __builtin_nontemporal_store(dv1, vp + 1);
            __builtin_nontemporal_store(dv2, vp + 2);
        }
    }
}

extern "C" void kernel_launch(void* const* d_in, const int* in_sizes, int n_in,
                              void* d_out, int out_size, void* d_ws, size_t ws_size,
                              hipStream_t stream) {
    (void)in_sizes; (void)n_in; (void)out_size; (void)d_ws; (void)ws_size;
    const float* pos  = (const float*)d_in[0];   // [16,256,3]
    const float* cell = (const float*)d_in[1];   // [16,3,3]

    float* dist_out = (float*)d_out;                                        // [B,N,M]
    float* dvec_out = dist_out + (size_t)BATCH * NATOM * MCAND;             // [B,N,M,3]
    int*   nimg     = (int*)(dvec_out + (size_t)BATCH * NATOM * MCAND * 3); // [B] int32

    zero_nimg_kernel<<<1, 64, 0, stream>>>(nimg);
    radius_graph_kernel<<<(BATCH * NATOM) / 16, NTHREADS, 0, stream>>>(
        pos, cell, dist_out, dvec_out, nimg);
}